// CascadedAttention_76836964925817
// MI455X (gfx1250) — compile-verified
//
#include <hip/hip_runtime.h>

// ---------------------------------------------------------------------------
// CascadedAttention collapsed form (softmax over size-1 axis == ones):
//   colsum = sum_t x[t]                  (D,)      -- constant
//   c      = Co @ colsum                 (V,)      -- constant
//   U[t]   = Uo @ x[t]                   (T,V)     -- GEMM, done with f32 WMMA
//   y[t]   = sigmoid(Wo@y[t-1] + U[t-1] + c),  y[-1]=0, U[-1]=0  -- sequential
// Wa/Ua/Va provably do not affect the output.
// ---------------------------------------------------------------------------

typedef float v2f __attribute__((ext_vector_type(2)));
typedef float v8f __attribute__((ext_vector_type(8)));

#define TT 2048
#define DD 1024
#define VV 28
#define BSTRIDE 32      // padded row stride for B in workspace

// ws (floats): [0,1024) colsum | [1024,1056) cvec | [2048, 2048+TT*32) B rows
#define WS_CVEC 1024
#define WS_B    2048

// ---- Kernel 1: column sums of x (streams the 8MB of x, warms L2) ----------
__global__ __launch_bounds__(256) void ca_colsum(const float* __restrict__ x,
                                                 float* __restrict__ ws) {
    int d = blockIdx.x * 256 + threadIdx.x;   // 0..1023, fully coalesced rows
    float s = 0.f;
    for (int t = 0; t < TT; ++t) s += x[t * DD + d];
    ws[d] = s;
}

// ---- Kernel 2: cvec = Co @ colsum; also B[0] = cvec -----------------------
__global__ __launch_bounds__(1024) void ca_cvec(const float* __restrict__ Co,
                                                float* __restrict__ ws) {
    const float* colsum = ws;
    float* cvec = ws + WS_CVEC;
    float* B0   = ws + WS_B;          // row t=0 of B
    int v    = threadIdx.x >> 5;      // one wave32 per output element
    int lane = threadIdx.x & 31;
    float s = 0.f;
    if (v < VV) {
        for (int k = lane; k < DD; k += 32) s += Co[v * DD + k] * colsum[k];
    }
    for (int off = 16; off > 0; off >>= 1) s += __shfl_down(s, off, 32);
    if (lane == 0) { cvec[v] = s; B0[v] = s; }   // v=28..31 write 0
}

// ---- Kernel 3: B[t+1] = Uo @ x[t] + cvec via V_WMMA_F32_16X16X4_F32 -------
// One wave handles 16 rows of t; K-loop 1024/4; two N-tiles cover V=28 (pad 32).
// A (16x4 f32) layout per ISA 7.12.2: lane l(0..15) row M=l, VGPR0=K(2*half),
// VGPR1=K(2*half+1). B (4x16) mirrors: VGPR0 rows K=2*half, VGPR1 K=2*half+1,
// N striped across lanes.
__global__ __launch_bounds__(32) void ca_uo_gemm(const float* __restrict__ x,
                                                 const float* __restrict__ Uo,
                                                 float* __restrict__ ws) {
    const float* cvec = ws + WS_CVEC;
    float* B = ws + WS_B;

    const int lane = threadIdx.x;
    const int half = lane >> 4;          // 0 or 1
    const int l    = lane & 15;
    const int t0   = blockIdx.x * 16;
    const int m    = t0 + l;             // A row (frame index)
    const int v1   = 16 + l;             // N-tile-1 column

    const float* aptr  = x  + m * DD + 2 * half;
    const float* b0ptr = Uo + l  * DD + 2 * half;                 // cols 0..15
    const float  m1    = (v1 < VV) ? 1.f : 0.f;                   // pad mask
    const float* b1ptr = Uo + (v1 < VV ? v1 : 0) * DD + 2 * half; // cols 16..27

    v8f acc0 = {}; v8f acc1 = {};
#pragma unroll 4
    for (int k = 0; k < DD; k += 4) {
        v2f a;  a.x  = aptr[k];        a.y  = aptr[k + 1];
        v2f b0; b0.x = b0ptr[k];       b0.y = b0ptr[k + 1];
        v2f b1; b1.x = b1ptr[k] * m1;  b1.y = b1ptr[k + 1] * m1;
        acc0 = __builtin_amdgcn_wmma_f32_16x16x4_f32(false, a, false, b0,
                                                     (short)0, acc0, false, false);
        acc1 = __builtin_amdgcn_wmma_f32_16x16x4_f32(false, a, false, b1,
                                                     (short)0, acc1, false, false);
    }

    // C/D layout: VGPR r -> (M = r + 8*half, N = l) ; tile1 N = 16 + l
    const float add0 = cvec[l];
    const float add1 = (v1 < VV) ? cvec[v1] : 0.f;
    const int mrow = t0 + 8 * half;
#pragma unroll
    for (int r = 0; r < 8; ++r) {
        const int tr = mrow + r + 1;               // shifted: B[t+1] = U[t]+c
        if (tr < TT) {
            B[tr * BSTRIDE + l] = acc0[r] + add0;
            if (v1 < VV) B[tr * BSTRIDE + v1] = acc1[r] + add1;
        }
    }
}

// ---- Kernel 4: sequential sigmoid recurrence, single wave -----------------
// Lane v owns y[v]; Wo row in 28 VGPRs; y broadcast with v_readlane (SGPR
// operand FMAs); 8-deep register prefetch hides the per-step L2 load of B[t].
__global__ __launch_bounds__(32) void ca_recur(const float* __restrict__ Wo,
                                               const float* __restrict__ ws,
                                               float* __restrict__ out) {
    const float* B = ws + WS_B;
    const int v = threadIdx.x;

    float w[VV];
#pragma unroll
    for (int j = 0; j < VV; ++j) w[j] = (v < VV) ? Wo[v * VV + j] : 0.f;

    constexpr int PF = 8;
    float buf[PF];
#pragma unroll
    for (int i = 0; i < PF; ++i) buf[i] = B[i * BSTRIDE + v];

    float y = 0.f;
    for (int tb = 0; tb < TT; tb += PF) {
#pragma unroll
        for (int i = 0; i < PF; ++i) {
            const int t = tb + i;
            const float bv = buf[i];
            if (t + PF < TT) buf[i] = B[(t + PF) * BSTRIDE + v];  // prefetch

            const int yi = __float_as_int(y);
            float s0 = bv, s1 = 0.f, s2 = 0.f, s3 = 0.f;
#pragma unroll
            for (int j = 0; j < VV; j += 4) {
                s0 = fmaf(w[j],     __int_as_float(__builtin_amdgcn_readlane(yi, j)),     s0);
                s1 = fmaf(w[j + 1], __int_as_float(__builtin_amdgcn_readlane(yi, j + 1)), s1);
                s2 = fmaf(w[j + 2], __int_as_float(__builtin_amdgcn_readlane(yi, j + 2)), s2);
                s3 = fmaf(w[j + 3], __int_as_float(__builtin_amdgcn_readlane(yi, j + 3)), s3);
            }
            const float s = (s0 + s1) + (s2 + s3);
            y = __builtin_amdgcn_rcpf(1.0f + __expf(-s));   // sigmoid
            if (v < VV) out[t * VV + v] = y;                // fire-and-forget
        }
    }
}

// ---------------------------------------------------------------------------
extern "C" void kernel_launch(void* const* d_in, const int* in_sizes, int n_in,
                              void* d_out, int out_size, void* d_ws, size_t ws_size,
                              hipStream_t stream) {
    (void)in_sizes; (void)n_in; (void)out_size; (void)ws_size;
    const float* x  = (const float*)d_in[0];
    // d_in[1]=Wa, d_in[2]=Ua, d_in[3]=Va: mathematically dead (softmax over size-1 axis)
    const float* Wo = (const float*)d_in[4];
    const float* Uo = (const float*)d_in[5];
    const float* Co = (const float*)d_in[6];
    float* out = (float*)d_out;
    float* ws  = (float*)d_ws;

    ca_colsum <<<DD / 256, 256, 0, stream>>>(x, ws);
    ca_cvec   <<<1, 1024, 0, stream>>>(Co, ws);
    ca_uo_gemm<<<TT / 16, 32, 0, stream>>>(x, Uo, ws);
    ca_recur  <<<1, 32, 0, stream>>>(Wo, ws, out);
}